// TensorProductInteractionBlock_48137993454065
// MI455X (gfx1250) — compile-verified
//
#include <hip/hip_runtime.h>
#include <math.h>

// TensorProductInteractionBlock for MI455X (gfx1250).
//   K0: zero message accumulator (N x 256 f32)
//   K1: node up-projection  (WMMA f32 16x16x4, weights staged in LDS)
//   K2: FUSED edge MLP (WMMA) + tensor-product message + f32 atomic scatter
//       (each wave processes 4 edge tiles to amortize the LDS weight stage)
//   K3: node out-projection (WMMA)
// WMMA A is a v2f from global/LDS (two strided loads in K1/K3); B is one b64 v2f load
// from a pair-interleaved XOR-swizzled LDS weight stage (bank-conflict free;
// compiler lowers pairs of them to ds_load_2addr_stride64_b64).

#define MUL 32
#define NN 50000
#define NE 800000
#define EDGE_TILES_PER_WAVE 4

typedef __attribute__((ext_vector_type(2))) float v2f;
typedef __attribute__((ext_vector_type(8))) float v8f;

__device__ __forceinline__ v8f wmma_f32(v2f a, v2f b, v8f c) {
  return __builtin_amdgcn_wmma_f32_16x16x4_f32(false, a, false, b, (short)0, c,
                                               false, false);
}

__device__ __forceinline__ float silu(float x) {
  return x / (1.0f + __expf(-x));
}

// Pair-interleaved swizzled weight stage.
// Element W[k][n] lives at  g*2N + 2*(n ^ ((g&1)<<4)) + (k&1),  g = k>>1.
// B-operand for K-rows {kb, kb+1} at column n is ONE v2f load; the g-parity
// XOR flips bit4 of n so the two lane halves hit disjoint 32-bank sets.
__device__ __forceinline__ void stage_w(float* dst, const float* __restrict__ src,
                                        int K, int N, int tid, int nthr) {
  for (int idx = tid; idx < K * N; idx += nthr) {
    const int k = idx / N;
    const int n = idx - k * N;
    const int g = k >> 1;
    dst[g * 2 * N + 2 * (n ^ ((g & 1) << 4)) + (k & 1)] = src[idx];
  }
}

__device__ __forceinline__ v2f ldb(const float* wl, int N2, int g, int n) {
  return *(const v2f*)&wl[g * N2 + 2 * (n ^ ((g & 1) << 4))];
}

// ---------------------------------------------------------------- K0: zero msg
__global__ void zero_kernel(float4* __restrict__ p, int n4) {
  int i = blockIdx.x * blockDim.x + threadIdx.x;
  if (i < n4) p[i] = make_float4(0.f, 0.f, 0.f, 0.f);
}

// ------------------------------------------------------- K1: up-projection
// 128 threads = 4 waves; wave sigma: 0 -> scalar channel, 1..3 -> vector comp.
// (16 nodes x 32) @ (32 x 32), 8 K-steps x 2 N-tiles. Branch-free addressing.
__global__ void __launch_bounds__(128) up_kernel(const float* __restrict__ nf,
                                                 const float* __restrict__ Wus,
                                                 const float* __restrict__ Wuv,
                                                 float* __restrict__ up) {
  __shared__ __align__(16) float wl[2 * 1024];  // 2 x (16 groups x 64)
  stage_w(wl, Wus, 32, 32, threadIdx.x, 128);
  stage_w(wl + 1024, Wuv, 32, 32, threadIdx.x, 128);
  __syncthreads();

  const int lane = threadIdx.x & 31;
  const int sigma = threadIdx.x >> 5;
  const int tile = blockIdx.x;
  const int m = lane & 15;
  const int half = lane >> 4;
  const float inv = 0.17677669529663689f;  // 1/sqrt(32)
  const int i = sigma - 1;
  const float* rowA = nf + (size_t)(tile * 16 + m) * 128;
  const float* abase = rowA + ((sigma == 0) ? 0 : 32 + i);
  const int astr = (sigma == 0) ? 1 : 3;
  const float* mywl = wl + ((sigma == 0) ? 0 : 1024);
  const int sbase = (sigma == 0) ? 0 : 32 + i;
  const int sstr = (sigma == 0) ? 1 : 3;

  v8f acc0 = {};
  v8f acc1 = {};
#pragma unroll
  for (int kk = 0; kk < 8; ++kk) {
    const int kb = kk * 4 + half * 2;
    const int g = kk * 2 + half;
    v2f a;
    a.x = abase[kb * astr];
    a.y = abase[(kb + 1) * astr];
    acc0 = wmma_f32(a, ldb(mywl, 64, g, m), acc0);
    acc1 = wmma_f32(a, ldb(mywl, 64, g, m + 16), acc1);
  }
#pragma unroll
  for (int r = 0; r < 8; ++r) {
    const size_t node = tile * 16 + r + half * 8;
    up[node * 128 + sbase + m * sstr] = acc0[r] * inv;
    up[node * 128 + sbase + (m + 16) * sstr] = acc1[r] * inv;
  }
}

// ------------------------------ K2: fused edge MLP + message + atomic scatter
// 128 threads = 4 waves; each wave runs EDGE_TILES_PER_WAVE 16-edge tiles
// (200 WMMA per tile) against the block-shared LDS weight stage.
// LDS: 51.2 KB weights + per-wave h (16x68) and tpw (16x132) -> 100 KB/block.
#define HS 68
#define TS 132
__global__ void __launch_bounds__(128) edge_kernel(
    const float* __restrict__ ef,   // (E,8)
    const float* __restrict__ ea,   // (E,4)
    const int* __restrict__ eidx,   // (2,E)
    const float* __restrict__ A1, const float* __restrict__ b1,
    const float* __restrict__ A2, const float* __restrict__ b2,
    const float* __restrict__ A3,
    const float* __restrict__ up,   // (N,128)
    float* __restrict__ msg)        // (N,256)
{
  __shared__ __align__(16) float w1l[8 * 64];    //  512 floats
  __shared__ __align__(16) float w2l[64 * 64];   // 4096 floats
  __shared__ __align__(16) float w3l[64 * 128];  // 8192 floats
  __shared__ __align__(16) float sh_h[4][16 * HS];
  __shared__ __align__(16) float sh_t[4][16 * TS];

  stage_w(w1l, A1, 8, 64, threadIdx.x, 128);
  stage_w(w2l, A2, 64, 64, threadIdx.x, 128);
  stage_w(w3l, A3, 64, 128, threadIdx.x, 128);
  __syncthreads();

  const int lane = threadIdx.x & 31;
  const int wv = threadIdx.x >> 5;
  const int m = lane & 15;
  const int half = lane >> 4;
  float* h = sh_h[wv];
  float* tp = sh_t[wv];
  const float invR3 = 0.57735026918962576f;  // 1/sqrt(3)
  const int u0 = half * 16;
  const int tile0 = (blockIdx.x * 4 + wv) * EDGE_TILES_PER_WAVE;

#pragma unroll 1
  for (int t = 0; t < EDGE_TILES_PER_WAVE; ++t) {
    const int e0 = (tile0 + t) * 16;

    // ---- layer 1: h = silu(ef(16x8) @ A1(8x64) + b1)
    v8f acc[4] = {};
#pragma unroll
    for (int kk = 0; kk < 2; ++kk) {
      const int kb = kk * 4 + half * 2;
      const int g = kk * 2 + half;
      v2f a = *(const v2f*)&ef[(size_t)(e0 + m) * 8 + kb];
#pragma unroll
      for (int nt = 0; nt < 4; ++nt)
        acc[nt] = wmma_f32(a, ldb(w1l, 128, g, m + nt * 16), acc[nt]);
    }
#pragma unroll
    for (int nt = 0; nt < 4; ++nt) {
      const int n = m + nt * 16;
      const float bias = b1[n];
#pragma unroll
      for (int r = 0; r < 8; ++r)
        h[(r + half * 8) * HS + n] = silu(acc[nt][r] + bias);
    }

    // ---- layer 2: h = silu(h(16x64) @ A2(64x64) + b2)
    v8f acc2[4] = {};
#pragma unroll
    for (int kk = 0; kk < 16; ++kk) {
      const int kb = kk * 4 + half * 2;
      const int g = kk * 2 + half;
      v2f a = *(const v2f*)&h[m * HS + kb];
#pragma unroll
      for (int nt = 0; nt < 4; ++nt)
        acc2[nt] = wmma_f32(a, ldb(w2l, 128, g, m + nt * 16), acc2[nt]);
    }
#pragma unroll
    for (int nt = 0; nt < 4; ++nt) {
      const int n = m + nt * 16;
      const float bias = b2[n];
#pragma unroll
      for (int r = 0; r < 8; ++r)
        h[(r + half * 8) * HS + n] = silu(acc2[nt][r] + bias);  // DS in-order
    }

    // ---- layer 3: tpw = h(16x64) @ A3(64x128), fold in the /10
    v8f acc3[8] = {};
#pragma unroll
    for (int kk = 0; kk < 16; ++kk) {
      const int kb = kk * 4 + half * 2;
      const int g = kk * 2 + half;
      v2f a = *(const v2f*)&h[m * HS + kb];
#pragma unroll
      for (int nt = 0; nt < 8; ++nt)
        acc3[nt] = wmma_f32(a, ldb(w3l, 256, g, m + nt * 16), acc3[nt]);
    }
#pragma unroll
    for (int nt = 0; nt < 8; ++nt) {
      const int n = m + nt * 16;
#pragma unroll
      for (int r = 0; r < 8; ++r)
        tp[(r + half * 8) * TS + n] = acc3[nt][r] * 0.1f;
    }

    // ---- tensor-product message + scatter (lane: edge m, 16 channels)
    const int eg = e0 + m;
    const int snd = eidx[eg];
    const int rcv = eidx[NE + eg];
    const float ys = ea[(size_t)eg * 4 + 0];
    const float yv0 = ea[(size_t)eg * 4 + 1];
    const float yv1 = ea[(size_t)eg * 4 + 2];
    const float yv2 = ea[(size_t)eg * 4 + 3];
    const float* us = up + (size_t)snd * 128;
    float* mr = msg + (size_t)rcv * 256;
#pragma unroll 4
    for (int du = 0; du < 16; ++du) {
      const int u = u0 + du;
      const float w0 = tp[m * TS + u];
      const float w1 = tp[m * TS + 32 + u];
      const float w2 = tp[m * TS + 64 + u];
      const float w3 = tp[m * TS + 96 + u];
      const float ssu = us[u];
      const float vx = us[32 + u * 3 + 0];
      const float vy = us[32 + u * 3 + 1];
      const float vz = us[32 + u * 3 + 2];
      atomicAdd(&mr[u], ssu * ys * w0);
      atomicAdd(&mr[32 + u], (vx * yv0 + vy * yv1 + vz * yv2) * invR3 * w3);
      atomicAdd(&mr[64 + u * 3 + 0], ssu * yv0 * w1);
      atomicAdd(&mr[64 + u * 3 + 1], ssu * yv1 * w1);
      atomicAdd(&mr[64 + u * 3 + 2], ssu * yv2 * w1);
      atomicAdd(&mr[160 + u * 3 + 0], vx * ys * w2);
      atomicAdd(&mr[160 + u * 3 + 1], vy * ys * w2);
      atomicAdd(&mr[160 + u * 3 + 2], vz * ys * w2);
    }
  }
}

// ------------------------------------------------------ K3: out-projection
// (16 nodes x 64) @ (64 x 32) per wave; sigma 0 -> scalar, 1..3 -> vector i.
__global__ void __launch_bounds__(128) out_kernel(const float* __restrict__ msg,
                                                  const float* __restrict__ Wos,
                                                  const float* __restrict__ Wov,
                                                  float* __restrict__ out) {
  __shared__ __align__(16) float wl[2 * 2048];  // 2 x (32 groups x 64)
  stage_w(wl, Wos, 64, 32, threadIdx.x, 128);
  stage_w(wl + 2048, Wov, 64, 32, threadIdx.x, 128);
  __syncthreads();

  const int lane = threadIdx.x & 31;
  const int sigma = threadIdx.x >> 5;
  const int tile = blockIdx.x;
  const int m = lane & 15;
  const int half = lane >> 4;
  const float inv2 = 0.125f;  // 1/sqrt(64)
  const int i = sigma - 1;
  const float* rowA = msg + (size_t)(tile * 16 + m) * 256;
  const float* abase = rowA + ((sigma == 0) ? 0 : 64 + i);
  const int astr = (sigma == 0) ? 1 : 3;
  const float* mywl = wl + ((sigma == 0) ? 0 : 2048);
  const int sbase = (sigma == 0) ? 0 : 32 + i;
  const int sstr = (sigma == 0) ? 1 : 3;

  v8f acc0 = {};
  v8f acc1 = {};
#pragma unroll
  for (int kk = 0; kk < 16; ++kk) {
    const int kb = kk * 4 + half * 2;
    const int g = kk * 2 + half;
    v2f a;
    a.x = abase[kb * astr];
    a.y = abase[(kb + 1) * astr];
    acc0 = wmma_f32(a, ldb(mywl, 64, g, m), acc0);
    acc1 = wmma_f32(a, ldb(mywl, 64, g, m + 16), acc1);
  }
#pragma unroll
  for (int r = 0; r < 8; ++r) {
    const size_t node = tile * 16 + r + half * 8;
    out[node * 128 + sbase + m * sstr] = acc0[r] * inv2;
    out[node * 128 + sbase + (m + 16) * sstr] = acc1[r] * inv2;
  }
}

extern "C" void kernel_launch(void* const* d_in, const int* in_sizes, int n_in,
                              void* d_out, int out_size, void* d_ws,
                              size_t ws_size, hipStream_t stream) {
  const float* nf  = (const float*)d_in[0];   // node_feats (N,128)
  const float* ea  = (const float*)d_in[1];   // edge_attrs (E,4)
  const float* ef  = (const float*)d_in[2];   // edge_feats (E,8)
  const int* eidx  = (const int*)d_in[3];     // edge_index (2,E)
  const float* Wus = (const float*)d_in[4];   // W_up_s (32,32)
  const float* Wuv = (const float*)d_in[5];   // W_up_v (32,32)
  const float* A1  = (const float*)d_in[6];   // (8,64)
  const float* b1  = (const float*)d_in[7];   // (64,)
  const float* A2  = (const float*)d_in[8];   // (64,64)
  const float* b2  = (const float*)d_in[9];   // (64,)
  const float* A3  = (const float*)d_in[10];  // (64,128)
  const float* Wos = (const float*)d_in[11];  // (64,32)
  const float* Wov = (const float*)d_in[12];  // (64,32)
  float* out = (float*)d_out;                 // (N,128)

  float* up  = (float*)d_ws;                  // (N,128) up-projected feats
  float* msg = up + (size_t)NN * 128;         // (N,256) message accumulator

  const int n4 = NN * 256 / 4;
  zero_kernel<<<(n4 + 255) / 256, 256, 0, stream>>>((float4*)msg, n4);
  up_kernel<<<NN / 16, 128, 0, stream>>>(nf, Wus, Wuv, up);
  edge_kernel<<<NE / 16 / 4 / EDGE_TILES_PER_WAVE, 128, 0, stream>>>(
      ef, ea, eidx, A1, b1, A2, b2, A3, up, msg);
  out_kernel<<<NN / 16, 128, 0, stream>>>(msg, Wos, Wov, out);
}